// RoIPool_54520314855547
// MI455X (gfx1250) — compile-verified
//
#include <hip/hip_runtime.h>

// RoIPool for MI455X (gfx1250).
// The ROI rectangle (rw x rh x CB channels, strides W and H*W) is a textbook
// Tensor Data Mover tile. Each block owns one ROI x 16 channels and pipelines
// eight 2-channel TDM tile loads (TENSORcnt-tracked, zero thread-level loads)
// against LDS max-pooling with a double buffer. The op is pure data movement
// (~17 MB HBM traffic at 23.3 TB/s; features are L2-resident), so the win is
// coalesced DMA staging + native DS reads; WMMA does not apply to max pooling.

#define CC   256
#define HH   64
#define WW   64
#define NROI 256
#define OH   7
#define OW   7
#define CB   2                    // channels per TDM tile -> 32 KB worst case
#define GRP  8                    // tiles (channel-groups) per block
#define TILE_ELEMS (CB * HH * WW) // 8192 floats = 32 KB
#define TILE_BYTES (TILE_ELEMS * 4)

typedef __attribute__((ext_vector_type(4))) unsigned int u32x4;
typedef __attribute__((ext_vector_type(8))) int          i32x8;
typedef __attribute__((ext_vector_type(4))) int          i32x4;

// Explicit LDS-address-space pointer so pooling reads lower to ds_load_b32
// (generic-pointer access was lowering to flat_load_b32).
typedef __attribute__((address_space(3))) const float lds_cfloat;

__device__ __forceinline__ static void
tdm_issue(unsigned long long ga, unsigned int lds_off, int rw, int rh)
{
    // D# group 0: count=1, lds_addr, global tile address, type=2
    u32x4 g0;
    g0[0] = 1u;
    g0[1] = lds_off;
    g0[2] = (unsigned int)ga;
    g0[3] = ((unsigned int)(ga >> 32) & 0x01FFFFFFu) | (2u << 30);

    // D# group 1 bitfields (ISA 8.4):
    //  [17:16] data_size=2 (4B)   [79:48] tensor_dim0=W
    //  [111:80] tensor_dim1=H     [127:112] tile_dim0=rw
    //  [143:128] tile_dim1=rh     [159:144] tile_dim2=CB
    //  [207:160] dim0_stride=W    [255:208] dim1_stride=H*W
    i32x8 g1;
    g1[0] = (int)(2u << 16);
    g1[1] = (int)(((unsigned)WW & 0xFFFFu) << 16);
    g1[2] = (int)((((unsigned)WW >> 16) & 0xFFFFu) |
                  (((unsigned)HH & 0xFFFFu) << 16));
    g1[3] = (int)((((unsigned)HH >> 16) & 0xFFFFu) |
                  (((unsigned)rw & 0xFFFFu) << 16));
    g1[4] = (int)(((unsigned)rh & 0xFFFFu) |
                  (((unsigned)CB & 0xFFFFu) << 16));
    g1[5] = (int)((unsigned)WW);                            // dim0_stride lo32
    g1[6] = (int)((((unsigned)(HH * WW)) & 0xFFFFu) << 16); // dim1_stride lo16
    g1[7] = (int)(((unsigned)(HH * WW)) >> 16);             // dim1_stride hi

    // D# group 2: tensor_dim2=C, no dim3, no iteration
    i32x4 g2;
    g2[0] = CC; g2[1] = 0; g2[2] = 0; g2[3] = 0;

    // D# group 3: unused (no dim3/dim4, not gather mode)
    i32x4 g3;
    g3[0] = 0; g3[1] = 0; g3[2] = 0; g3[3] = 0;

    // Extra descriptor words required by this toolchain's 6-arg builtin.
    i32x8 gx;
    gx[0] = 0; gx[1] = 0; gx[2] = 0; gx[3] = 0;
    gx[4] = 0; gx[5] = 0; gx[6] = 0; gx[7] = 0;

    __builtin_amdgcn_tensor_load_to_lds(g0, g1, g2, g3, gx, 0);
}

extern "C" __global__ __launch_bounds__(128)
void roipool_tdm_kernel(const float* __restrict__ features,
                        const float* __restrict__ rois,
                        float* __restrict__ out)
{
    __shared__ float tile[2 * TILE_ELEMS];   // 64 KB double buffer

    // Wave-relative LDS byte offset of the tile: flat addresses in the LDS
    // aperture carry the LDS offset in bits [31:0] (ISA 10.2).
    const unsigned int tile_lds_base =
        (unsigned int)(unsigned long long)(const void*)&tile[0];
    lds_cfloat* ltile = (lds_cfloat*)&tile[0];

    const int n     = blockIdx.y;              // ROI index
    const int cbase = blockIdx.x * (CB * GRP); // first channel of this block

    // ---- ROI -> integer bounds (exactly the reference semantics) ----
    const float rx1 = rois[4 * n + 0];
    const float ry1 = rois[4 * n + 1];
    const float rx2 = rois[4 * n + 2];
    const float ry2 = rois[4 * n + 3];

    int x1 = (int)(rx1 * (float)WW);
    int y1 = (int)(ry1 * (float)HH);
    int x2 = (int)(rx2 * (float)WW);
    int y2 = (int)(ry2 * (float)HH);
    x1 = min(max(x1, 0), WW - 1);
    x2 = max(x1 + 1, min(x2, WW));
    y1 = min(max(y1, 0), HH - 1);
    y2 = max(y1 + 1, min(y2, HH));

    // Values are block-uniform; pin them to SGPRs (TDM descriptor needs SGPRs).
    x1 = __builtin_amdgcn_readfirstlane(x1);
    y1 = __builtin_amdgcn_readfirstlane(y1);
    x2 = __builtin_amdgcn_readfirstlane(x2);
    y2 = __builtin_amdgcn_readfirstlane(y2);

    const int rw = x2 - x1;   // 1..64
    const int rh = y2 - y1;   // 1..64

    const unsigned long long ga0 =
        (unsigned long long)(const void*)features +
        4ull * ((unsigned long long)cbase * (HH * WW) +
                (unsigned long long)y1 * WW +
                (unsigned long long)x1);
    const unsigned long long ga_step = 4ull * CB * HH * WW; // next channel pair

    const bool issuer = (threadIdx.x < 32);   // wave 0 drives the TDM

    // ---- per-thread pooling geometry (fixed across groups) ----
    const int t       = threadIdx.x;
    const bool pooler = (t < CB * OH * OW);
    const int cl   = t / (OH * OW);
    const int cell = t - cl * (OH * OW);
    const int oh   = cell / OW;
    const int ow   = cell - oh * OW;
    // reference bin bounds (relative to tile origin)
    const int lsh = (oh * rh) / OH;
    const int leh = ((oh + 1) * rh + (OH - 1)) / OH;
    const int lsw = (ow * rw) / OW;
    const int lew = ((ow + 1) * rw + (OW - 1)) / OW;
    const int chan_base = cl * (rh * rw);     // packed tile: rw*rh per channel

    float* outp = out + ((((long)n * CC) + cbase + cl) * OH + oh) * OW + ow;

    // ---- software pipeline: TDM(g+1) overlaps pooling of tile g ----
    if (issuer)
        tdm_issue(ga0, tile_lds_base, rw, rh);

    for (int g = 0; g < GRP; ++g) {
        const int buf = g & 1;

        if (issuer) {
            if (g + 1 < GRP) {
                tdm_issue(ga0 + (unsigned long long)(g + 1) * ga_step,
                          tile_lds_base + (unsigned int)((g + 1) & 1) * TILE_BYTES,
                          rw, rh);
                __builtin_amdgcn_s_wait_tensorcnt(1);  // tile g complete
            } else {
                __builtin_amdgcn_s_wait_tensorcnt(0);  // last tile complete
            }
        }
        __syncthreads();   // publish tile g to all 4 waves

        if (pooler) {
            const int base = buf * TILE_ELEMS + chan_base;
            float m = -__builtin_huge_valf();
            for (int y = lsh; y < leh; ++y) {
                const int rowb = base + y * rw;
                for (int x = lsw; x < lew; ++x)
                    m = fmaxf(m, ltile[rowb + x]);   // addrspace(3) -> ds_load_b32
            }
            outp[(long)g * (CB * OH * OW)] = m;
        }
        __syncthreads();   // tile[buf] free before TDM(g+2) overwrites it
    }
}

extern "C" void kernel_launch(void* const* d_in, const int* in_sizes, int n_in,
                              void* d_out, int out_size, void* d_ws, size_t ws_size,
                              hipStream_t stream)
{
    const float* features = (const float*)d_in[0];   // [256, 64, 64] f32
    const float* rois     = (const float*)d_in[1];   // [256, 4] f32
    float*       out      = (float*)d_out;           // [256, 256, 7, 7] f32

    dim3 grid(CC / (CB * GRP), NROI);                // (16, 256) blocks
    roipool_tdm_kernel<<<grid, 128, 0, stream>>>(features, rois, out);
}